// FocalLoss_67001489817982
// MI455X (gfx1250) — compile-verified
//
#include <hip/hip_runtime.h>
#include <math.h>

typedef __attribute__((ext_vector_type(2))) float v2f;
typedef __attribute__((ext_vector_type(8))) float v8f;

#define B_N 8
#define A_N 65536
#define C_N 80
#define M_N 32
#define BDIM 256

__device__ __forceinline__ float iou_box(float a0, float a1, float a2, float a3,
                                         float b0, float b1, float b2, float b3) {
    float area_a = (a2 - a0) * (a3 - a1);
    float area_b = (b2 - b0) * (b3 - b1);
    float iw = fmaxf(fminf(a2, b2) - fmaxf(a0, b0), 0.0f);
    float ih = fmaxf(fminf(a3, b3) - fmaxf(a1, b1), 0.0f);
    float inter = iw * ih;
    return inter / fmaxf(area_a + area_b - inter, 1e-8f);
}

// Wave32 reduction via V_WMMA_F32_16X16X4_F32.
// A (16x4): lane L supplies a.x (K=0 or 2) and a.y (K=1 or 3); this packs the
// 32 lanes' (x,y) pairs into one A matrix. B (4x16) is a 0/1 selector so that
// columns N<8 sum the x-slots and columns N>=8 sum the y-slots. After the MMA,
// lane 0 + lane 16 hold the 16 row-contributions of column 0 (sum = total x),
// lanes 8 + 24 hold column 8 (sum = total y). Two atomics per wave per value.
__device__ __forceinline__ void wave_red2_wmma(float x, float y, float* outx, float* outy) {
    int lane = threadIdx.x & 31;
    v2f a; a.x = x; a.y = y;
    float lo = ((lane & 15) < 8) ? 1.0f : 0.0f;
    v2f bsel; bsel.x = lo; bsel.y = 1.0f - lo;
    v8f c = {};
    c = __builtin_amdgcn_wmma_f32_16x16x4_f32(
        /*neg_a=*/false, a, /*neg_b=*/false, bsel,
        /*c_mod=*/(short)0, c, /*reuse_a=*/false, /*reuse_b=*/false);
    float s = ((c[0] + c[1]) + (c[2] + c[3])) + ((c[4] + c[5]) + (c[6] + c[7]));
    if (lane == 0 || lane == 16) atomicAdd(outx, s);
    if (lane == 8 || lane == 24) atomicAdd(outy, s);
}

__global__ void init_ws_kernel(float* f, int nf, int* gi, int ni) {
    int t = blockIdx.x * blockDim.x + threadIdx.x;
    if (t < nf) f[t] = 0.0f;
    if (t < ni) gi[t] = 0;
}

__global__ void __launch_bounds__(BDIM) pass1_kernel(
        const float* __restrict__ cls, const float* __restrict__ reg,
        const float* __restrict__ anchors, const float* __restrict__ ann,
        float* cls_sum, float* reg_sum, float* num_pos, int* gt_used) {
    __shared__ float s_ann[M_N * 5];
    int gid = blockIdx.x * BDIM + threadIdx.x;
    int b = gid >> 16;            // A_N == 65536
    int a = gid & (A_N - 1);
    if (threadIdx.x < M_N * 5) s_ann[threadIdx.x] = ann[b * M_N * 5 + threadIdx.x];
    __syncthreads();

    float4 an = ((const float4*)anchors)[a];
    float aw = an.z - an.x, ah = an.w - an.y;
    float acx = an.x + 0.5f * aw, acy = an.y + 0.5f * ah;

    float best = -3.0f; int barg = 0;
    for (int m = 0; m < M_N; ++m) {
        float lbl = s_ann[m * 5 + 4];
        float iou = (lbl != -1.0f)
            ? iou_box(an.x, an.y, an.z, an.w,
                      s_ann[m * 5 + 0], s_ann[m * 5 + 1], s_ann[m * 5 + 2], s_ann[m * 5 + 3])
            : -1.0f;
        if (iou > best) { best = iou; barg = m; }   // strict > keeps first argmax
    }
    bool pos = best >= 0.5f;
    bool neg = best < 0.4f;

    float cls_part = 0.0f;
    if (pos || neg) {
        int lbl = pos ? (int)s_ann[barg * 5 + 4] : -1;
        const float4* cp = (const float4*)(cls + (size_t)gid * C_N);
        #pragma unroll
        for (int c4 = 0; c4 < C_N / 4; ++c4) {
            float4 v = cp[c4];
            float pv[4] = {v.x, v.y, v.z, v.w};
            #pragma unroll
            for (int j = 0; j < 4; ++j) {
                int cc = c4 * 4 + j;
                float p = fminf(fmaxf(pv[j], 1e-4f), 1.0f - 1e-4f);
                if (cc == lbl) cls_part += 0.25f * (1.0f - p) * (1.0f - p) * (-__logf(p));
                else           cls_part += 0.75f * p * p * (-__logf(1.0f - p));
            }
        }
    }

    float reg_part = 0.0f;
    if (pos) {
        atomicOr(&gt_used[b * M_N + barg], 1);
        float4 rp = ((const float4*)reg)[gid];
        float g0 = s_ann[barg * 5 + 0], g1 = s_ann[barg * 5 + 1];
        float g2 = s_ann[barg * 5 + 2], g3 = s_ann[barg * 5 + 3];
        float gwr = g2 - g0, ghr = g3 - g1;
        float gcx = g0 + 0.5f * gwr, gcy = g1 + 0.5f * ghr;
        float gw = fmaxf(gwr, 1.0f), gh = fmaxf(ghr, 1.0f);
        float t0 = ((gcx - acx) / aw) * 10.0f;   // / STD 0.1
        float t1 = ((gcy - acy) / ah) * 10.0f;
        float t2 = __logf(gw / aw) * 5.0f;       // / STD 0.2
        float t3 = __logf(gh / ah) * 5.0f;
        float d[4] = {fabsf(t0 - rp.x), fabsf(t1 - rp.y), fabsf(t2 - rp.z), fabsf(t3 - rp.w)};
        #pragma unroll
        for (int j = 0; j < 4; ++j)
            reg_part += (d[j] <= (1.0f / 9.0f)) ? 4.5f * d[j] * d[j] : d[j] - (0.5f / 9.0f);
    }

    // full-wave WMMA reductions (all 32 lanes of a wave share b)
    wave_red2_wmma(cls_part, reg_part, &cls_sum[b], &reg_sum[b]);
    wave_red2_wmma(pos ? 1.0f : 0.0f, 0.0f, &num_pos[b], &reg_sum[b]); // y adds 0: harmless
}

__global__ void __launch_bounds__(BDIM) pass2_kernel(
        const float* __restrict__ reg, const float* __restrict__ loc,
        const float* __restrict__ anchors, const float* __restrict__ ann,
        const int* __restrict__ gt_used, float* loc_sum) {
    __shared__ float s_ann[M_N * 5];
    __shared__ int s_used[M_N];
    int gid = blockIdx.x * BDIM + threadIdx.x;
    int b = gid >> 16;
    int a = gid & (A_N - 1);
    if (threadIdx.x < M_N * 5) s_ann[threadIdx.x] = ann[b * M_N * 5 + threadIdx.x];
    if (threadIdx.x < M_N) s_used[threadIdx.x] = gt_used[b * M_N + threadIdx.x];
    __syncthreads();

    float4 an = ((const float4*)anchors)[a];
    float aw = an.z - an.x, ah = an.w - an.y;
    float acx = an.x + 0.5f * aw, acy = an.y + 0.5f * ah;

    float best = -3.0f;
    for (int m = 0; m < M_N; ++m) {
        float lbl = s_ann[m * 5 + 4];
        float iou = (lbl != -1.0f)
            ? iou_box(an.x, an.y, an.z, an.w,
                      s_ann[m * 5 + 0], s_ann[m * 5 + 1], s_ann[m * 5 + 2], s_ann[m * 5 + 3])
            : -1.0f;
        best = fmaxf(best, iou);
    }
    bool pos = best >= 0.5f;

    float loc_part = 0.0f;
    if (pos) {
        float4 rp = ((const float4*)reg)[gid];
        float pcx = acx + rp.x * 0.1f * aw;
        float pcy = acy + rp.y * 0.1f * ah;
        float pw = __expf(rp.z * 0.2f) * aw;
        float ph = __expf(rp.w * 0.2f) * ah;
        float s0 = fminf(fmaxf(pcx - 0.5f * pw, 0.0f), 512.0f);
        float s1 = fminf(fmaxf(pcy - 0.5f * ph, 0.0f), 512.0f);
        float s2 = fminf(fmaxf(pcx + 0.5f * pw, 0.0f), 512.0f);
        float s3 = fminf(fmaxf(pcy + 0.5f * ph, 0.0f), 512.0f);
        float ism = -1e9f;
        for (int m = 0; m < M_N; ++m) {
            if (s_used[m]) {
                float iou = iou_box(s0, s1, s2, s3,
                                    s_ann[m * 5 + 0], s_ann[m * 5 + 1],
                                    s_ann[m * 5 + 2], s_ann[m * 5 + 3]);
                ism = fmaxf(ism, iou);
            }
        }
        float ls = fminf(fmaxf(1.0f - fabsf(loc[gid] - ism), 1e-4f), 1.0f - 1e-4f);
        loc_part = -__logf(ls);
    }
    wave_red2_wmma(loc_part, 0.0f, &loc_sum[b], &loc_sum[b]); // y adds 0: harmless
}

__global__ void finalize_kernel(const float* __restrict__ ann,
                                const float* cls_sum, const float* reg_sum,
                                const float* loc_sum, const float* num_pos, float* out) {
    __shared__ float s_c[B_N], s_r[B_N], s_l[B_N];
    int t = threadIdx.x;
    if (t < B_N) {
        bool has_gt = false;
        for (int m = 0; m < M_N; ++m) has_gt |= (ann[t * M_N * 5 + m * 5 + 4] != -1.0f);
        float np = num_pos[t];
        bool okpos = has_gt && (np > 0.0f);
        s_c[t] = has_gt ? cls_sum[t] / fmaxf(np, 1.0f) : 0.0f;
        s_r[t] = okpos ? reg_sum[t] / fmaxf(np * 4.0f, 1.0f) : 0.0f;
        s_l[t] = okpos ? loc_sum[t] / fmaxf(np, 1.0f) : 0.0f;
    }
    __syncthreads();
    if (t == 0) {
        float c = 0.0f, r = 0.0f, l = 0.0f;
        for (int b = 0; b < B_N; ++b) { c += s_c[b]; r += s_r[b]; l += s_l[b]; }
        out[0] = c / (float)B_N;
        out[1] = r / (float)B_N;
        out[2] = l / (float)B_N;
    }
}

extern "C" void kernel_launch(void* const* d_in, const int* in_sizes, int n_in,
                              void* d_out, int out_size, void* d_ws, size_t ws_size,
                              hipStream_t stream) {
    const float* cls     = (const float*)d_in[0]; // [B,A,C]
    const float* reg     = (const float*)d_in[1]; // [B,A,4]
    const float* loc     = (const float*)d_in[2]; // [B,A,1]
    const float* anchors = (const float*)d_in[3]; // [1,A,4]
    const float* ann     = (const float*)d_in[4]; // [B,M,5]
    float* out = (float*)d_out;

    float* ws = (float*)d_ws;
    float* cls_sum = ws;
    float* reg_sum = ws + B_N;
    float* loc_sum = ws + 2 * B_N;
    float* num_pos = ws + 3 * B_N;
    int*   gt_used = (int*)(ws + 4 * B_N);

    init_ws_kernel<<<1, 256, 0, stream>>>(ws, 4 * B_N, gt_used, B_N * M_N);

    int total = B_N * A_N;
    pass1_kernel<<<total / BDIM, BDIM, 0, stream>>>(cls, reg, anchors, ann,
                                                    cls_sum, reg_sum, num_pos, gt_used);
    pass2_kernel<<<total / BDIM, BDIM, 0, stream>>>(reg, loc, anchors, ann,
                                                    gt_used, loc_sum);
    finalize_kernel<<<1, 32, 0, stream>>>(ann, cls_sum, reg_sum, loc_sum, num_pos, out);
}